// UIG_3770981285908
// MI455X (gfx1250) — compile-verified
//
#include <hip/hip_runtime.h>
#include <hip/hip_bf16.h>
#include <math.h>

// Reference fixes these sizes.
#define N_USERS_C 16384
#define BLOCK 128
#define CAP 1024   // per-row LDS staging capacity (mean segment length ~61)

#if __has_builtin(__builtin_amdgcn_global_load_async_to_lds_b32)
#define HAVE_ASYNC_LDS 1
#else
#define HAVE_ASYNC_LDS 0
#endif

// Typed address-space pointers matching the builtin's signature:
//   __builtin_amdgcn_global_load_async_to_lds_b32(global int*, local int*, imm, imm)
typedef __attribute__((address_space(1))) int* g_i32p;
typedef __attribute__((address_space(3))) int* l_i32p;

// row_ptr[r] = first index k with rows[k] >= r  (rows sorted ascending)
__global__ void UIG_row_ptr_kernel(const int* __restrict__ rows,
                                   int* __restrict__ row_ptr,
                                   int nnz, int n_users) {
  int r = blockIdx.x * blockDim.x + threadIdx.x;
  if (r > n_users) return;
  int lo = 0, hi = nnz;
  while (lo < hi) {
    int mid = (lo + hi) >> 1;
    if (rows[mid] < r) lo = mid + 1; else hi = mid;
  }
  row_ptr[r] = lo;
}

// For entry k in a row segment: a = R[r,c] (sum of dups of c), b = R[r,c^1]
__device__ __forceinline__ void pair_sums(const int* __restrict__ cseg,
                                          const float* __restrict__ vseg,
                                          int L, int k, float& pop, float& nic) {
  const int c = cseg[k];
  const int cp = c ^ 1;
  float a = 0.f, b = 0.f;
  for (int j = 0; j < L; ++j) {
    const int cj = cseg[j];
    const float vj = vseg[j];
    a += (cj == c) ? vj : 0.f;
    b += (cj == cp) ? vj : 0.f;
  }
  pop = 0.5f * (a + b);
  nic = 0.5f * (a - b);
}

__device__ __forceinline__ float block_sum(float v, float* s_red) {
  const int tid = threadIdx.x;
  s_red[tid] = v;
  __syncthreads();
  for (int off = BLOCK >> 1; off > 0; off >>= 1) {
    if (tid < off) s_red[tid] += s_red[tid + off];
    __syncthreads();
  }
  float r = s_red[0];
  __syncthreads();
  return r;
}

__device__ __forceinline__ float block_max(float v, float* s_red) {
  const int tid = threadIdx.x;
  s_red[tid] = v;
  __syncthreads();
  for (int off = BLOCK >> 1; off > 0; off >>= 1) {
    if (tid < off) s_red[tid] = fmaxf(s_red[tid], s_red[tid + off]);
    __syncthreads();
  }
  float r = s_red[0];
  __syncthreads();
  return r;
}

__global__ __launch_bounds__(BLOCK)
void UIG_segment_kernel(const float* __restrict__ vals,
                        const int* __restrict__ cols,
                        const int* __restrict__ row_ptr,
                        float* __restrict__ out, int nnz) {
  const int r = blockIdx.x;
  const int start = row_ptr[r];
  const int end = row_ptr[r + 1];
  const int L = end - start;
  if (L <= 0) return;
  const int tid = threadIdx.x;

  __shared__ int   s_cols[CAP];
  __shared__ float s_vals[CAP];
  __shared__ float s_pop[CAP];
  __shared__ float s_nic[CAP];
  __shared__ float s_red[BLOCK];

  const bool fits = (L <= CAP);

  if (tid == 0) {
    __builtin_prefetch(cols + start, 0, 0);   // global_prefetch_b8
    __builtin_prefetch(vals + start, 0, 0);
  }

  if (fits) {
#if HAVE_ASYNC_LDS
    // CDNA5 async global -> LDS copy, tracked by ASYNCcnt (no VGPR round trip)
    for (int k = tid; k < L; k += BLOCK) {
      __builtin_amdgcn_global_load_async_to_lds_b32(
          (g_i32p)(void*)(cols + start + k),
          (l_i32p)(void*)(&s_cols[k]), 0, 0);
      __builtin_amdgcn_global_load_async_to_lds_b32(
          (g_i32p)(void*)(vals + start + k),
          (l_i32p)(void*)(&s_vals[k]), 0, 0);
    }
#if __has_builtin(__builtin_amdgcn_s_wait_asynccnt)
    __builtin_amdgcn_s_wait_asynccnt(0);
#else
    asm volatile("s_wait_asynccnt 0" ::: "memory");
#endif
#else
    for (int k = tid; k < L; k += BLOCK) {
      s_cols[k] = cols[start + k];
      s_vals[k] = vals[start + k];
    }
#endif
  }
  __syncthreads();

  // Phase 1: per-entry pop/nic + per-thread partial sum/max
  float psum_pop = 0.f, pmax_pop = -__builtin_inff();
  float psum_nic = 0.f, pmax_nic = -__builtin_inff();
  for (int k = tid; k < L; k += BLOCK) {
    float pop, nic;
    if (fits) pair_sums(s_cols, s_vals, L, k, pop, nic);
    else      pair_sums(cols + start, vals + start, L, k, pop, nic);
    if (fits) { s_pop[k] = pop; s_nic[k] = nic; }
    psum_pop += pop; pmax_pop = fmaxf(pmax_pop, pop);
    psum_nic += nic; pmax_nic = fmaxf(pmax_nic, nic);
  }
  __syncthreads();

  const float sum_pop = block_sum(psum_pop, s_red);
  const float max_pop = block_max(pmax_pop, s_red);
  const float sum_nic = block_sum(psum_nic, s_red);
  const float max_nic = block_max(pmax_nic, s_red);

  // Phase 2: softmax denominators
  float pd_pop = 0.f, pd_nic = 0.f;
  for (int k = tid; k < L; k += BLOCK) {
    float pop, nic;
    if (fits) { pop = s_pop[k]; nic = s_nic[k]; }
    else      pair_sums(cols + start, vals + start, L, k, pop, nic);
    pd_pop += expf(pop - max_pop);
    pd_nic += expf(nic - max_nic);
  }
  const float denom_pop = block_sum(pd_pop, s_red);
  const float denom_nic = block_sum(pd_nic, s_red);

  // Phase 3: outputs (softmax only if segment sum of original values > 0)
  const bool ap = sum_pop > 0.f;
  const bool an = sum_nic > 0.f;
  for (int k = tid; k < L; k += BLOCK) {
    float pop, nic;
    if (fits) { pop = s_pop[k]; nic = s_nic[k]; }
    else      pair_sums(cols + start, vals + start, L, k, pop, nic);
    out[start + k]       = ap ? (expf(pop - max_pop) / denom_pop) : pop;
    out[nnz + start + k] = an ? (expf(nic - max_nic) / denom_nic) : nic;
  }
}

extern "C" void kernel_launch(void* const* d_in, const int* in_sizes, int n_in,
                              void* d_out, int out_size, void* d_ws, size_t ws_size,
                              hipStream_t stream) {
  const float* values = (const float*)d_in[0];
  const int*   rows   = (const int*)d_in[1];
  const int*   cols   = (const int*)d_in[2];
  const int    nnz    = in_sizes[0];
  const int    n_users = N_USERS_C;   // fixed by the reference problem

  int* row_ptr = (int*)d_ws;          // (n_users + 1) ints of scratch

  const int t1 = 256;
  const int g1 = (n_users + 1 + t1 - 1) / t1;
  UIG_row_ptr_kernel<<<g1, t1, 0, stream>>>(rows, row_ptr, nnz, n_users);
  UIG_segment_kernel<<<n_users, BLOCK, 0, stream>>>(values, cols, row_ptr,
                                                    (float*)d_out, nnz);
}